// Model_73117523247742
// MI455X (gfx1250) — compile-verified
//
#include <hip/hip_runtime.h>

// Ragged segment-mean embedding (EmbeddingBag 'mean') + positional add,
// reformulated as   out[seg,:] = (H[seg,:64] @ E[64,1024]) / max(cnt,1) + pos[seg%512,:]
// where H is the per-segment token-count histogram (exact small ints in fp32).
// Executed with chained V_WMMA_F32_16X16X4_F32 (fp32 A/B/C: matches fp32 reference).
// Output (134 MB, write-once) is streamed with non-temporal stores to keep the
// L2 free for the L2-resident embedding/positional tables.

typedef float v2f __attribute__((ext_vector_type(2)));
typedef float v8f __attribute__((ext_vector_type(8)));

constexpr int T_TOK   = 131072;       // total ragged tokens
constexpr int VOCAB   = 64;
constexpr int D_MODEL = 1024;
constexpr int S_LEN   = 512;
constexpr int N_SEG   = 64 * 512;     // B*S = 32768 segments
constexpr int SEG_PER_WG = 64;        // 4 waves x 16-row M-tiles
constexpr int HPITCH  = 66;           // pad 64->66 floats: breaks LDS bank conflicts

__global__ __launch_bounds__(128)
void phon_embed_bag_wmma(const int*   __restrict__ tokens,     // [T]
                         const int*   __restrict__ seg_ids,    // [T], sorted
                         const float* __restrict__ emb,        // [64][1024]
                         const float* __restrict__ pos,        // [512][1024]
                         float*       __restrict__ out)        // [32768][1024]
{
    __shared__ float Hs[SEG_PER_WG][HPITCH];   // per-segment token histogram (A matrix)
    __shared__ int   bounds[SEG_PER_WG + 1];
    __shared__ float cnts[SEG_PER_WG];

    const int tid     = threadIdx.x;
    const int segBase = blockIdx.x * SEG_PER_WG;

    // ---- zero histogram ----
    for (int i = tid; i < SEG_PER_WG * HPITCH; i += 128)
        (&Hs[0][0])[i] = 0.0f;

    // ---- lower_bound of each segment boundary in the sorted segment_ids ----
    if (tid <= SEG_PER_WG) {
        const int target = segBase + tid;
        int lo = 0, hi = T_TOK;
        while (lo < hi) {
            int mid = (lo + hi) >> 1;
            if (seg_ids[mid] < target) lo = mid + 1; else hi = mid;
        }
        bounds[tid] = lo;
    }
    __syncthreads();

    // ---- each thread owns one segment row: private histogram, no atomics ----
    if (tid < SEG_PER_WG) {
        const int b0 = bounds[tid], b1 = bounds[tid + 1];
        for (int i = b0; i < b1; ++i)
            Hs[tid][tokens[i]] += 1.0f;
        cnts[tid] = (float)(b1 - b0);
    }
    __syncthreads();

    // ---- WMMA GEMM: D_tile(16x16) = A(16x64) x B(64x16), chained K=4 steps ----
    const int lane  = tid & 31;
    const int wave  = tid >> 5;
    const int mrow  = wave * 16;                 // wave's M-tile inside the 64-seg block
    const int l15   = lane & 15;
    const int koff  = (lane >= 16) ? 2 : 0;      // A/B f32 layout: half-waves hold K pairs
    const int mhalf = (lane >= 16) ? 8 : 0;      // C/D layout: VGPR r <-> rows r / r+8

    // A fragments held in registers across the whole N loop (reuse 16x)
    v2f a[16];
#pragma unroll
    for (int k = 0; k < 16; ++k) {
        a[k].x = Hs[mrow + l15][4 * k + koff];
        a[k].y = Hs[mrow + l15][4 * k + koff + 1];
    }

    // per-output-row reciprocal counts (mean), empty segments -> divide by 1
    float rcp[8];
#pragma unroll
    for (int r = 0; r < 8; ++r)
        rcp[r] = 1.0f / fmaxf(cnts[mrow + mhalf + r], 1.0f);

    const int segRow0  = segBase + mrow;         // first output row of this wave's tile
    const int posRow0  = segRow0 % S_LEN;        // 16-aligned; never wraps inside a tile
    const int colBase  = blockIdx.y * (16 * 16); // 16 N-tiles per y-slice

    for (int nt = 0; nt < 16; ++nt) {
        const int n0  = colBase + nt * 16;
        const int col = n0 + l15;

        v8f acc = {};
#pragma unroll
        for (int k = 0; k < 16; ++k) {
            v2f b;                               // B 4x16 f32: rows K=4k+koff(+1), cols=lanes
            b.x = emb[(4 * k + koff)     * D_MODEL + col];
            b.y = emb[(4 * k + koff + 1) * D_MODEL + col];
            acc = __builtin_amdgcn_wmma_f32_16x16x4_f32(
                      /*neg_a=*/false, a[k], /*neg_b=*/false, b,
                      /*c_mod=*/(short)0, acc, /*reuse_a=*/false, /*reuse_b=*/false);
        }

        // epilogue: mean + positional add; coalesced, non-temporal streaming stores
#pragma unroll
        for (int r = 0; r < 8; ++r) {
            const int m = mhalf + r;
            const float v = acc[r] * rcp[r] + pos[(posRow0 + m) * D_MODEL + col];
            __builtin_nontemporal_store(v, &out[(size_t)(segRow0 + m) * D_MODEL + col]);
        }
    }
}

extern "C" void kernel_launch(void* const* d_in, const int* in_sizes, int n_in,
                              void* d_out, int out_size, void* d_ws, size_t ws_size,
                              hipStream_t stream) {
    const int*   tokens = (const int*)  d_in[0];
    const int*   segids = (const int*)  d_in[1];
    const float* emb    = (const float*)d_in[2];
    const float* pos    = (const float*)d_in[3];
    float*       out    = (float*)      d_out;

    dim3 grid(N_SEG / SEG_PER_WG,          // 512 M-blocks (64 segments each)
              D_MODEL / 256);              // 4 N-slices of 256 columns
    phon_embed_bag_wmma<<<grid, 128, 0, stream>>>(tokens, segids, emb, pos, out);
}